// GPRORouter_89472758710467
// MI455X (gfx1250) — compile-verified
//
#include <hip/hip_runtime.h>
#include <hip/hip_bf16.h>
#include <math.h>

// ---------------------------------------------------------------------------
// GPRO router, MI455X (gfx1250, wave32).
// Memory-bound (X is L2-resident, ~0.7GB HBM traffic @ 23.3TB/s >> matrix
// time), so full fp32 precision via V_WMMA_F32_16X16X4_F32.
// This revision: K-pair-interleaved B tiles (ds_load_b64 fragments, no
// register-packing movs) + GLOBAL_LOAD_ASYNC_TO_LDS_B128 staging for the
// A (activation) tiles, synchronized with s_wait_asynccnt.
// ---------------------------------------------------------------------------

typedef float v2f __attribute__((ext_vector_type(2)));
typedef float v8f __attribute__((ext_vector_type(8)));

#define D_DIM   2048
#define E_DIM   64
#define M_TOK   16384          // B*S = 4*4096

#define TILE_M  128
#define TILE_N  128
#define TILE_K  16
#define LDA     20             // As row stride (floats; pad, multiple of 4)
#define LDB1    288            // Bi kpair stride gemm1: 128*2 + 32  (%64 == 32)
#define LDB2    160            // Bi kpair stride gemm2: 64*2 + 32   (%64 == 32)

__device__ __forceinline__ float gelu_exact(float v) {
    return 0.5f * v * (1.0f + erff(v * 0.70710678118654752f));
}

// Async copy 16B global -> LDS (per-lane LDS byte address), ASYNCcnt-tracked.
__device__ __forceinline__ void async_load_b128(unsigned lds_byte_addr,
                                                const float* gptr) {
    asm volatile("global_load_async_to_lds_b128 %0, %1, off"
                 :: "v"(lds_byte_addr), "v"(gptr) : "memory");
}
__device__ __forceinline__ void wait_async0() {
    asm volatile("s_wait_asynccnt 0x0" ::: "memory");
}

// ======================= GEMM1: H = gelu(X @ W1 + b1) ======================
// grid = (N/128, M/128, 2)   z=0: router weights -> Hr, z=1: baseline -> Hb
__global__ __launch_bounds__(256) void gemm1_gelu(
    const float* __restrict__ X,
    const float* __restrict__ W1r, const float* __restrict__ b1r,
    const float* __restrict__ W1b, const float* __restrict__ b1b,
    float* __restrict__ Hr, float* __restrict__ Hb)
{
    const float* W    = (blockIdx.z == 0) ? W1r : W1b;
    const float* bias = (blockIdx.z == 0) ? b1r : b1b;
    float*       H    = (blockIdx.z == 0) ? Hr  : Hb;

    __shared__ float As[TILE_M][LDA];   // 128 x 16 fp32 (async-staged)
    __shared__ float Bi[8][LDB1];       // Bi[p][c*2+s] = W[k0+2p+s][nTile+c]

    const int tid    = threadIdx.x;
    const int lane   = tid & 31;
    const int wave   = tid >> 5;
    const int wm     = wave & 3;        // 4 waves down M
    const int wn     = wave >> 2;       // 2 waves across N
    const int lrow16 = lane & 15;
    const int khalf  = lane >> 4;       // lanes 16-31: K+2 / M+8 halves

    const int mTile = blockIdx.y * TILE_M;
    const int nTile = blockIdx.x * TILE_N;

    const unsigned asBase = (unsigned)(uintptr_t)&As[0][0];

    v8f acc[2][4] = {};                 // wave tile 32(M) x 64(N)

    for (int k0 = 0; k0 < D_DIM; k0 += TILE_K) {
        // --- A tile: 128x16 = 512 x b128, 2 async copies per thread ---
        #pragma unroll
        for (int i = 0; i < 2; ++i) {
            int idx = tid + i * 256;    // 0..511
            int r   = idx >> 2;
            int c4  = (idx & 3) * 4;
            async_load_b128(asBase + (unsigned)(r * LDA + c4) * 4u,
                            &X[(size_t)(mTile + r) * D_DIM + k0 + c4]);
        }
        // --- B tile: K-pair interleaved, 1024 float2, 4 per thread ---
        #pragma unroll
        for (int i = 0; i < 4; ++i) {
            int idx = tid + i * 256;    // 0..1023
            int p   = idx >> 7;         // kpair 0..7
            int c   = idx & 127;
            float2 v;
            v.x = W[(size_t)(k0 + 2 * p    ) * D_DIM + nTile + c];
            v.y = W[(size_t)(k0 + 2 * p + 1) * D_DIM + nTile + c];
            *(float2*)&Bi[p][c * 2] = v;
        }
        if (k0 + TILE_K < D_DIM)
            __builtin_prefetch(&W[(size_t)(k0 + TILE_K + (tid >> 4)) * D_DIM + nTile], 0, 0);
        wait_async0();
        __syncthreads();

        #pragma unroll
        for (int kk = 0; kk < 4; ++kk) {       // 4 x (K=4) = K-step 16
            v2f a[2], b[4];
            #pragma unroll
            for (int mf = 0; mf < 2; ++mf) {
                int row = wm * 32 + mf * 16 + lrow16;
                a[mf] = *(const v2f*)&As[row][kk * 4 + khalf * 2];
            }
            #pragma unroll
            for (int nf = 0; nf < 4; ++nf) {
                int col = wn * 64 + nf * 16 + lrow16;
                b[nf] = *(const v2f*)&Bi[kk * 2 + khalf][col * 2];
            }
            #pragma unroll
            for (int mf = 0; mf < 2; ++mf)
                #pragma unroll
                for (int nf = 0; nf < 4; ++nf)
                    acc[mf][nf] = __builtin_amdgcn_wmma_f32_16x16x4_f32(
                        false, a[mf], false, b[nf], (short)0, acc[mf][nf], false, false);
        }
        __syncthreads();
    }

    // epilogue: bias + exact GELU, fp32 store
    #pragma unroll
    for (int mf = 0; mf < 2; ++mf)
        #pragma unroll
        for (int nf = 0; nf < 4; ++nf)
            #pragma unroll
            for (int r = 0; r < 8; ++r) {
                int grow = mTile + wm * 32 + mf * 16 + r + khalf * 8;
                int gcol = nTile + wn * 64 + nf * 16 + lrow16;
                float v = acc[mf][nf][r] + bias[gcol];
                H[(size_t)grow * D_DIM + gcol] = gelu_exact(v);
            }
}

// ================= GEMM2: scores = Hr @ W2r + b2r  (N = 64) ================
// grid = (M/128)
__global__ __launch_bounds__(256) void gemm2_scores(
    const float* __restrict__ Hr, const float* __restrict__ W2r,
    const float* __restrict__ b2r, float* __restrict__ scores)
{
    __shared__ float As[TILE_M][LDA];   // 128 x 16
    __shared__ float Bi[8][LDB2];       // Bi[p][c*2+s] = W2r[k0+2p+s][c]

    const int tid    = threadIdx.x;
    const int lane   = tid & 31;
    const int wave   = tid >> 5;        // 8 waves, 16 rows each
    const int lrow16 = lane & 15;
    const int khalf  = lane >> 4;
    const int mTile  = blockIdx.x * TILE_M;

    const unsigned asBase = (unsigned)(uintptr_t)&As[0][0];

    v8f acc[4] = {};                    // wave tile 16(M) x 64(N)

    for (int k0 = 0; k0 < D_DIM; k0 += TILE_K) {
        #pragma unroll
        for (int i = 0; i < 2; ++i) {
            int idx = tid + i * 256;
            int r   = idx >> 2;
            int c4  = (idx & 3) * 4;
            async_load_b128(asBase + (unsigned)(r * LDA + c4) * 4u,
                            &Hr[(size_t)(mTile + r) * D_DIM + k0 + c4]);
        }
        #pragma unroll
        for (int i = 0; i < 2; ++i) {   // 8 x 64 = 512 float2, 2 per thread
            int idx = tid + i * 256;
            int p   = idx >> 6;
            int c   = idx & 63;
            float2 v;
            v.x = W2r[(size_t)(k0 + 2 * p    ) * E_DIM + c];
            v.y = W2r[(size_t)(k0 + 2 * p + 1) * E_DIM + c];
            *(float2*)&Bi[p][c * 2] = v;
        }
        wait_async0();
        __syncthreads();

        #pragma unroll
        for (int kk = 0; kk < 4; ++kk) {
            v2f a, b[4];
            a = *(const v2f*)&As[wave * 16 + lrow16][kk * 4 + khalf * 2];
            #pragma unroll
            for (int nf = 0; nf < 4; ++nf)
                b[nf] = *(const v2f*)&Bi[kk * 2 + khalf][(nf * 16 + lrow16) * 2];
            #pragma unroll
            for (int nf = 0; nf < 4; ++nf)
                acc[nf] = __builtin_amdgcn_wmma_f32_16x16x4_f32(
                    false, a, false, b[nf], (short)0, acc[nf], false, false);
        }
        __syncthreads();
    }

    #pragma unroll
    for (int nf = 0; nf < 4; ++nf)
        #pragma unroll
        for (int r = 0; r < 8; ++r) {
            int grow = mTile + wave * 16 + r + khalf * 8;
            int gcol = nf * 16 + lrow16;
            scores[(size_t)grow * E_DIM + gcol] = acc[nf][r] + b2r[gcol];
        }
}

// ================== baseline GEMV: b[t] = Hb[t,:] . W2b + b2b ==============
// grid = M/8, wave per token
__global__ __launch_bounds__(256) void baseline_kernel(
    const float* __restrict__ Hb, const float* __restrict__ W2b,
    const float* __restrict__ b2b, float* __restrict__ baseline)
{
    const int lane = threadIdx.x & 31;
    const int wave = threadIdx.x >> 5;
    const int t    = blockIdx.x * 8 + wave;

    const float4* H4 = (const float4*)(Hb + (size_t)t * D_DIM);
    const float4* W4 = (const float4*)W2b;
    float s = 0.0f;
    #pragma unroll
    for (int j = 0; j < 16; ++j) {      // 2048 floats = 512 float4 / 32 lanes
        float4 h = H4[lane + j * 32];
        float4 w = W4[lane + j * 32];
        s += h.x * w.x + h.y * w.y + h.z * w.z + h.w * w.w;
    }
    #pragma unroll
    for (int off = 16; off; off >>= 1) s += __shfl_xor(s, off, 32);
    if (lane == 0) baseline[t] = s + b2b[0];
}

// ============ finalize: gumbel + softmax(E=64) + top-2 + pg ================
// grid = M/8, wave per token; 2 experts per lane
__global__ __launch_bounds__(256) void finalize_kernel(
    const float* __restrict__ scores, const float* __restrict__ u,
    const float* __restrict__ baseline,
    float* __restrict__ outW, float* __restrict__ outI, float* __restrict__ outPG)
{
    const int lane = threadIdx.x & 31;
    const int wave = threadIdx.x >> 5;
    const int t    = blockIdx.x * 8 + wave;

    const int e0 = lane, e1 = lane + 32;
    float z0 = scores[(size_t)t * E_DIM + e0] - logf(-logf(u[(size_t)t * E_DIM + e0]));
    float z1 = scores[(size_t)t * E_DIM + e1] - logf(-logf(u[(size_t)t * E_DIM + e1]));

    float m = fmaxf(z0, z1);
    #pragma unroll
    for (int off = 16; off; off >>= 1) m = fmaxf(m, __shfl_xor(m, off, 32));
    float p0 = __expf(z0 - m), p1 = __expf(z1 - m);
    float sum = p0 + p1;
    #pragma unroll
    for (int off = 16; off; off >>= 1) sum += __shfl_xor(sum, off, 32);
    float inv = 1.0f / sum;
    p0 *= inv; p1 *= inv;

    // top-1 (ties -> lower index, matching lax.top_k)
    float bv; int bi;
    if (p0 >= p1) { bv = p0; bi = e0; } else { bv = p1; bi = e1; }
    #pragma unroll
    for (int off = 16; off; off >>= 1) {
        float ov = __shfl_xor(bv, off, 32);
        int   oi = __shfl_xor(bi, off, 32);
        if (ov > bv || (ov == bv && oi < bi)) { bv = ov; bi = oi; }
    }
    const int top1 = bi; const float w1 = bv;

    // top-2: mask out top1
    float c0 = (e0 == top1) ? -1.0f : p0;
    float c1 = (e1 == top1) ? -1.0f : p1;
    if (c0 >= c1) { bv = c0; bi = e0; } else { bv = c1; bi = e1; }
    #pragma unroll
    for (int off = 16; off; off >>= 1) {
        float ov = __shfl_xor(bv, off, 32);
        int   oi = __shfl_xor(bi, off, 32);
        if (ov > bv || (ov == bv && oi < bi)) { bv = ov; bi = oi; }
    }
    const int top2 = bi; const float w2 = bv;

    if (lane == 0) {
        const float base = baseline[t];
        outW [t * 2 + 0] = w1;            outW [t * 2 + 1] = w2;
        outI [t * 2 + 0] = (float)top1;   outI [t * 2 + 1] = (float)top2;
        outPG[t * 2 + 0] = w1 - base;     outPG[t * 2 + 1] = w2 - base;
    }
}

// ===========================================================================
extern "C" void kernel_launch(void* const* d_in, const int* in_sizes, int n_in,
                              void* d_out, int out_size, void* d_ws, size_t ws_size,
                              hipStream_t stream) {
    const float* x   = (const float*)d_in[0];   // [4,4096,2048]
    const float* u   = (const float*)d_in[1];   // [4,4096,64]
    const float* W1r = (const float*)d_in[2];
    const float* b1r = (const float*)d_in[3];
    const float* W2r = (const float*)d_in[4];
    const float* b2r = (const float*)d_in[5];
    const float* W1b = (const float*)d_in[6];
    const float* b1b = (const float*)d_in[7];
    const float* W2b = (const float*)d_in[8];
    const float* b2b = (const float*)d_in[9];

    float* out   = (float*)d_out;
    float* outW  = out;                                   // [M,2]
    float* outI  = out + (size_t)M_TOK * 2;               // [M,2]
    float* outB  = out + (size_t)M_TOK * 4;               // [M]
    float* outPG = out + (size_t)M_TOK * 5;               // [M,2]
    float* outS  = out + (size_t)M_TOK * 7;               // [M,64]

    float* Hr = (float*)d_ws;                             // [M, D] fp32
    float* Hb = Hr + (size_t)M_TOK * D_DIM;               // [M, D] fp32

    gemm1_gelu   <<<dim3(D_DIM / TILE_N, M_TOK / TILE_M, 2), 256, 0, stream>>>(
        x, W1r, b1r, W1b, b1b, Hr, Hb);
    gemm2_scores <<<dim3(M_TOK / TILE_M), 256, 0, stream>>>(Hr, W2r, b2r, outS);
    baseline_kernel<<<dim3(M_TOK / 8), 256, 0, stream>>>(Hb, W2b, b2b, outB);
    finalize_kernel<<<dim3(M_TOK / 8), 256, 0, stream>>>(outS, u, outB, outW, outI, outPG);
}